// GraphormerAttentionHead_57655640982217
// MI455X (gfx1250) — compile-verified
//
#include <hip/hip_runtime.h>
#include <hip/hip_bf16.h>
#include <math.h>

// ---------------- problem constants (match reference) ----------------
#define NN        4096
#define DIM_IN    128
#define DIM_QK    64
#define MAXDIST   5.0f
#define CH        16                      // column chunks for split flash attention
#define NTILES    (NN / 16)               // 256 row tiles
#define BLK_PER_CHUNK (NN / 16 / CH)      // 16 column blocks per chunk

typedef float v2f __attribute__((ext_vector_type(2)));
typedef float v8f __attribute__((ext_vector_type(8)));

// V_WMMA_F32_16X16X4_F32 : D(16x16 f32) = A(16x4 f32) x B(4x16 f32) + C
static __device__ __forceinline__ v8f wmma_f32_k4(v2f a, v2f b, v8f c) {
  return __builtin_amdgcn_wmma_f32_16x16x4_f32(
      /*neg_a=*/false, a, /*neg_b=*/false, b,
      /*c_mod=*/(short)0, c, /*reuse_a=*/false, /*reuse_b=*/false);
}

// ---------------------------------------------------------------------
// Kernel 1: csum[row] = sum_j nan_to_num( min(weights[row][j],5) * mean(ev) )
// ---------------------------------------------------------------------
__global__ void csum_kernel(const float* __restrict__ weights,
                            const float* __restrict__ edge_vector,
                            int ev_n,
                            float* __restrict__ csum) {
  __shared__ float red[256];
  const int row = blockIdx.x;
  const int tid = threadIdx.x;

  float evm = 0.f;
  for (int i = 0; i < ev_n; ++i) evm += edge_vector[i];
  evm /= (float)ev_n;

  float acc = 0.f;
  const float* wr = weights + (size_t)row * NN;
  for (int j = tid; j < NN; j += 256) {
    float w = fminf(wr[j], MAXDIST);
    float v = w * evm;
    if (isnan(v)) v = 0.f;
    else if (isinf(v)) v = copysignf(3.402823466e38f, v);
    acc += v;
  }
  red[tid] = acc;
  __syncthreads();
  for (int st = 128; st > 0; st >>= 1) {
    if (tid < st) red[tid] += red[tid + st];
    __syncthreads();
  }
  if (tid == 0) csum[row] = red[0];
}

// ---------------------------------------------------------------------
// Kernel 2: Q/K/V = x @ W + bias via fp32 WMMA. One wave -> one 16x16 tile.
// ---------------------------------------------------------------------
__global__ void qkv_kernel(const float* __restrict__ x,
                           const float* __restrict__ Wq, const float* __restrict__ bq,
                           const float* __restrict__ Wk, const float* __restrict__ bk,
                           const float* __restrict__ Wv, const float* __restrict__ bv,
                           float* __restrict__ qkv) {
  const int gw   = (blockIdx.x * blockDim.x + threadIdx.x) >> 5;
  const int lane = threadIdx.x & 31;
  const int lo   = lane & 15;
  const int hi   = lane >> 4;

  const int mat   = gw >> 10;          // 0=Q 1=K 2=V
  const int rem   = gw & 1023;
  const int m0    = (rem >> 2) * 16;
  const int n0    = (rem & 3) * 16;

  const float* W    = (mat == 0) ? Wq : (mat == 1) ? Wk : Wv;
  const float* bias = (mat == 0) ? bq : (mat == 1) ? bk : bv;
  float* out        = qkv + (size_t)mat * NN * DIM_QK;

  v8f acc = {};
  const float* arow = x + (size_t)(m0 + lo) * DIM_IN;
#pragma unroll
  for (int kk = 0; kk < DIM_IN / 4; ++kk) {
    const int k = kk * 4 + 2 * hi;
    v2f a = { arow[k], arow[k + 1] };
    const float* bp = W + (size_t)k * DIM_QK + n0 + lo;
    v2f b = { bp[0], bp[DIM_QK] };
    acc = wmma_f32_k4(a, b, acc);
  }

  const float bb = bias[n0 + lo];
#pragma unroll
  for (int r = 0; r < 8; ++r) {
    const int row = m0 + r + 8 * hi;
    out[(size_t)row * DIM_QK + n0 + lo] = acc[r] + bb;
  }
}

// ---------------------------------------------------------------------
// Kernel 3: split-column flash attention (partials). One wave owns
// (16-row tile, 256-column chunk); 4096 waves total for MLP/occupancy.
// Emits unnormalized O_partial plus per-row (m, l).
// ---------------------------------------------------------------------
__global__ void __launch_bounds__(256)
attn_partial_kernel(const float* __restrict__ Q, const float* __restrict__ K,
                    const float* __restrict__ V, const float* __restrict__ bmat,
                    const float* __restrict__ csum, const int* __restrict__ ptr,
                    int nptr,
                    float* __restrict__ part_O, float* __restrict__ part_m,
                    float* __restrict__ part_l) {
  __shared__ float pbuf[8][16][17];   // per-wave P tile (padded)

  const int wave  = threadIdx.x >> 5;
  const int lane  = threadIdx.x & 31;
  const int lo    = lane & 15;
  const int hi    = lane >> 4;
  const int gw    = blockIdx.x * 8 + wave;       // 0 .. NTILES*CH-1
  const int tile  = gw / CH;
  const int chunk = gw % CH;
  const int m0    = tile * 16;

  int rowg = 0;
  for (int t = 0; t < nptr; ++t) rowg += (ptr[t] <= m0) ? 1 : 0;

  // preload Q fragments (A layout)
  v2f qa[DIM_QK / 4];
  const float* qrow = Q + (size_t)(m0 + lo) * DIM_QK;
#pragma unroll
  for (int kk = 0; kk < DIM_QK / 4; ++kk) {
    const int k = kk * 4 + 2 * hi;
    qa[kk] = (v2f){ qrow[k], qrow[k + 1] };
  }

  float mrow[8], lrow[8];
  v8f accO[4] = {};
#pragma unroll
  for (int r = 0; r < 8; ++r) { mrow[r] = -3.0e38f; lrow[r] = 0.f; }

  const int jb0 = chunk * BLK_PER_CHUNK;
  for (int jb = jb0; jb < jb0 + BLK_PER_CHUNK; ++jb) {
    const int j0 = jb * 16;
    int colg = 0;
    for (int t = 0; t < nptr; ++t) colg += (ptr[t] <= j0) ? 1 : 0;
    const bool use_c = (colg == rowg);   // wave-uniform

    // ---- S = Q @ K^T (16 WMMA, K-dim 64)
    v8f s = {};
    const float* krow = K + (size_t)(j0 + lo) * DIM_QK;
#pragma unroll
    for (int kk = 0; kk < DIM_QK / 4; ++kk) {
      const int k = kk * 4 + 2 * hi;
      v2f kb = (v2f){ krow[k], krow[k + 1] };
      s = wmma_f32_k4(qa[kk], kb, s);
    }

    // ---- scores = s/8 + b + csum[col]*mask
    const float cadd = use_c ? csum[j0 + lo] : 0.f;
    float pr[8];
#pragma unroll
    for (int r = 0; r < 8; ++r) {
      const int row = m0 + r + 8 * hi;
      pr[r] = s[r] * 0.125f + bmat[(size_t)row * NN + j0 + lo] + cadd;
    }

    // ---- online softmax (row spans one 16-lane group)
#pragma unroll
    for (int r = 0; r < 8; ++r) {
      float rm = pr[r];
      rm = fmaxf(rm, __shfl_xor(rm, 1));
      rm = fmaxf(rm, __shfl_xor(rm, 2));
      rm = fmaxf(rm, __shfl_xor(rm, 4));
      rm = fmaxf(rm, __shfl_xor(rm, 8));
      const float mn = fmaxf(mrow[r], rm);
      const float sc = __expf(mrow[r] - mn);
      float p  = __expf(pr[r] - mn);
      float rs = p;
      rs += __shfl_xor(rs, 1);
      rs += __shfl_xor(rs, 2);
      rs += __shfl_xor(rs, 4);
      rs += __shfl_xor(rs, 8);
      lrow[r] = lrow[r] * sc + rs;
      mrow[r] = mn;
      pr[r]   = p;
#pragma unroll
      for (int nt = 0; nt < 4; ++nt) accO[nt][r] *= sc;
    }

    // ---- transpose P through wave-local LDS (C/D layout -> A layout)
#pragma unroll
    for (int r = 0; r < 8; ++r) pbuf[wave][r + 8 * hi][lo] = pr[r];

    // ---- O += P @ V_block (16 WMMA)
#pragma unroll
    for (int kk = 0; kk < 4; ++kk) {
      const int k = kk * 4 + 2 * hi;
      v2f pa = (v2f){ pbuf[wave][lo][k], pbuf[wave][lo][k + 1] };
      const float* vrow = V + (size_t)(j0 + k) * DIM_QK;
#pragma unroll
      for (int nt = 0; nt < 4; ++nt) {
        const float* vp = vrow + nt * 16 + lo;
        v2f vb = (v2f){ vp[0], vp[DIM_QK] };
        accO[nt] = wmma_f32_k4(pa, vb, accO[nt]);
      }
    }
  }

  // ---- emit partials (unnormalized O, row stats)
  float* po = part_O + (size_t)gw * 16 * DIM_QK;
#pragma unroll
  for (int nt = 0; nt < 4; ++nt) {
#pragma unroll
    for (int r = 0; r < 8; ++r) {
      po[(size_t)(r + 8 * hi) * DIM_QK + nt * 16 + lo] = accO[nt][r];
    }
  }
  if (lo == 0) {
#pragma unroll
    for (int r = 0; r < 8; ++r) {
      part_m[(size_t)gw * 16 + r + 8 * hi] = mrow[r];
      part_l[(size_t)gw * 16 + r + 8 * hi] = lrow[r];
    }
  }
}

// ---------------------------------------------------------------------
// Kernel 4: merge CH partials per row:  out = sum_c e^{m_c-m_g} O_c / l_g
// One thread per (row, d) element.
// ---------------------------------------------------------------------
__global__ void attn_merge_kernel(const float* __restrict__ part_O,
                                  const float* __restrict__ part_m,
                                  const float* __restrict__ part_l,
                                  float* __restrict__ out) {
  const int idx = blockIdx.x * blockDim.x + threadIdx.x;   // 0..NN*DIM_QK
  const int row = idx >> 6;
  const int d   = idx & (DIM_QK - 1);
  const int tile = row >> 4;
  const int rl   = row & 15;

  float mg = -3.0e38f;
#pragma unroll
  for (int c = 0; c < CH; ++c)
    mg = fmaxf(mg, part_m[(size_t)(tile * CH + c) * 16 + rl]);

  float lg = 0.f, o = 0.f;
#pragma unroll
  for (int c = 0; c < CH; ++c) {
    const size_t pc = (size_t)(tile * CH + c);
    const float e = __expf(part_m[pc * 16 + rl] - mg);
    lg += part_l[pc * 16 + rl] * e;
    o  += part_O[pc * 16 * DIM_QK + (size_t)rl * DIM_QK + d] * e;
  }
  out[idx] = o / lg;
}

// ---------------------------------------------------------------------
extern "C" void kernel_launch(void* const* d_in, const int* in_sizes, int n_in,
                              void* d_out, int out_size, void* d_ws, size_t ws_size,
                              hipStream_t stream) {
  const float* x       = (const float*)d_in[0];
  // d_in[1] = edge_attr (unused by the reference math)
  const float* bmat    = (const float*)d_in[2];
  const float* weights = (const float*)d_in[3];
  const int*   ptr     = (const int*)d_in[4];
  const float* Wq      = (const float*)d_in[5];
  const float* bq      = (const float*)d_in[6];
  const float* Wk      = (const float*)d_in[7];
  const float* bk      = (const float*)d_in[8];
  const float* Wv      = (const float*)d_in[9];
  const float* bv      = (const float*)d_in[10];
  const float* ev      = (const float*)d_in[11];

  float* ws     = (float*)d_ws;
  float* Qb     = ws;
  float* Kb     = Qb + (size_t)NN * DIM_QK;
  float* Vb     = Kb + (size_t)NN * DIM_QK;
  float* csum   = Vb + (size_t)NN * DIM_QK;
  float* part_m = csum + NN;
  float* part_l = part_m + (size_t)NTILES * CH * 16;
  float* part_O = part_l + (size_t)NTILES * CH * 16;
  // total ws use: 3*NN*64 + NN + 2*NTILES*CH*16 + NTILES*CH*16*64 floats ~ 20 MB

  csum_kernel<<<NN, 256, 0, stream>>>(weights, ev, in_sizes[11], csum);
  qkv_kernel<<<384, 256, 0, stream>>>(x, Wq, bq, Wk, bk, Wv, bv, ws);
  attn_partial_kernel<<<NTILES * CH / 8, 256, 0, stream>>>(
      Qb, Kb, Vb, bmat, csum, ptr, in_sizes[4], part_O, part_m, part_l);
  attn_merge_kernel<<<NN * DIM_QK / 256, 256, 0, stream>>>(part_O, part_m,
                                                           part_l, (float*)d_out);
}